// GCN_63513976373416
// MI455X (gfx1250) — compile-verified
//
#include <hip/hip_runtime.h>

// GCN (3x GraphConv, norm='both') for MI455X / gfx1250.
// feat: [N,128] f32; src/dst: [E] i32; W*: [128,128] f32 (row-major K x N); b*: [128] f32.

typedef __attribute__((ext_vector_type(2))) float v2f;
typedef __attribute__((ext_vector_type(8))) float v8f;

#define D 128

// ---------------- utility: zero fill ----------------
__global__ void gcn_zero_kernel(float* __restrict__ p, long long n) {
  long long i = (long long)blockIdx.x * blockDim.x + threadIdx.x;
  long long stride = (long long)gridDim.x * blockDim.x;
  for (; i < n; i += stride) p[i] = 0.0f;
}

// ---------------- degrees via float atomics ----------------
__global__ void gcn_degree_kernel(const int* __restrict__ src,
                                  const int* __restrict__ dst,
                                  float* __restrict__ deg_out,
                                  float* __restrict__ deg_in,
                                  int n_edges) {
  int e = blockIdx.x * blockDim.x + threadIdx.x;
  if (e >= n_edges) return;
  atomicAdd(&deg_out[src[e]], 1.0f);
  atomicAdd(&deg_in[dst[e]], 1.0f);
}

// deg -> rsqrt(max(deg, 1))   (in place; applied to deg_out||deg_in as one array)
__global__ void gcn_invsqrt_kernel(float* __restrict__ deg, int n) {
  int i = blockIdx.x * blockDim.x + threadIdx.x;
  if (i >= n) return;
  deg[i] = rsqrtf(fmaxf(deg[i], 1.0f));
}

// ---------------- SpMM: agg[dst] += h[src] * inv_out[src] ----------------
// 32 lanes per edge; each lane handles 4 consecutive dims (float4 gather, 4 f32 atomics).
__global__ __launch_bounds__(256) void gcn_spmm_kernel(
    const float* __restrict__ h, const int* __restrict__ src,
    const int* __restrict__ dst, const float* __restrict__ inv_out,
    float* __restrict__ agg, int n_edges) {
  int t = blockIdx.x * blockDim.x + threadIdx.x;
  int e = t >> 5;
  if (e >= n_edges) return;
  int lane = t & 31;
  int s = src[e];
  int d = dst[e];
  float sc = inv_out[s];
  const float4 v = *(const float4*)(h + (size_t)s * D + lane * 4);
  float* ap = agg + (size_t)d * D + lane * 4;
  atomicAdd(ap + 0, v.x * sc);
  atomicAdd(ap + 1, v.y * sc);
  atomicAdd(ap + 2, v.z * sc);
  atomicAdd(ap + 3, v.w * sc);
}

// ---------------- fused (inv_in scale) x GEMM x (+bias) x (ReLU) ----------------
// out[row0:row0+16, :] = relu( (inv_in[r] * agg[r,:]) @ W + b )
// One block = 16 output rows; 8 waves, wave w owns columns [16w, 16w+16).
// K = 128 consumed as 32 chained v_wmma_f32_16x16x4_f32.
//
// VGPR layouts per CDNA5 ISA 7.12.2 (fp32):
//   A 16x4 : lanes 0-15 -> K half {0,1}, lanes 16-31 -> K half {2,3}; vgpr = k within half; M = lane&15
//   B 4x16 : vgpr v, lane half hh -> row K = v + 2*hh ; N = lane&15
//   C/D    : vgpr r -> M = r + 8*(lane>=16) ; N = lane&15
__global__ __launch_bounds__(256) void gcn_gemm_kernel(
    const float* __restrict__ agg, const float* __restrict__ inv_in,
    const float* __restrict__ W, const float* __restrict__ bias,
    float* __restrict__ out, int relu) {
  __shared__ float Atile[16 * D];  // 8 KB, inv_in-scaled A tile

  const int row0 = blockIdx.x * 16;
  const int tid = threadIdx.x;

  // Cooperative scaled load of the 16x128 A tile: 256 threads x 8 floats.
  {
    int r = tid >> 4;            // 0..15
    int c = (tid & 15) * 8;      // 0,8,...,120
    float s = inv_in[row0 + r];
    const float4* g = (const float4*)(agg + (size_t)(row0 + r) * D + c);
    float4 v0 = g[0];
    float4 v1 = g[1];
    float4* l = (float4*)(Atile + r * D + c);
    v0.x *= s; v0.y *= s; v0.z *= s; v0.w *= s;
    v1.x *= s; v1.y *= s; v1.z *= s; v1.w *= s;
    l[0] = v0;
    l[1] = v1;
  }
  __syncthreads();

  const int lane = tid & 31;
  const int col0 = (tid >> 5) * 16;   // wave id * 16
  const int m = lane & 15;            // row-within-tile / n-within-tile
  const int khalf = (lane >> 4) * 2;  // 0 or 2

  v8f c = {};
#pragma unroll 4
  for (int k0 = 0; k0 < D; k0 += 4) {
    v2f a;
    a.x = Atile[m * D + k0 + khalf + 0];
    a.y = Atile[m * D + k0 + khalf + 1];
    v2f b;
    b.x = W[(size_t)(k0 + khalf + 0) * D + col0 + m];
    b.y = W[(size_t)(k0 + khalf + 1) * D + col0 + m];
    c = __builtin_amdgcn_wmma_f32_16x16x4_f32(
        /*neg_a=*/false, a, /*neg_b=*/false, b,
        /*c_mod=*/(short)0, c, /*reuse_a=*/false, /*reuse_b=*/false);
  }

  const int ncol = col0 + m;
  const int rbase = row0 + (lane >> 4) * 8;
  const float bv = bias[ncol];
#pragma unroll
  for (int r = 0; r < 8; ++r) {
    float v = c[r] + bv;
    if (relu) v = fmaxf(v, 0.0f);
    out[(size_t)(rbase + r) * D + ncol] = v;
  }
}

extern "C" void kernel_launch(void* const* d_in, const int* in_sizes, int n_in,
                              void* d_out, int out_size, void* d_ws, size_t ws_size,
                              hipStream_t stream) {
  const float* feat = (const float*)d_in[0];
  const int* src = (const int*)d_in[1];
  const int* dst = (const int*)d_in[2];
  const float* W1 = (const float*)d_in[3];
  const float* b1 = (const float*)d_in[4];
  const float* W2 = (const float*)d_in[5];
  const float* b2 = (const float*)d_in[6];
  const float* W3 = (const float*)d_in[7];
  const float* b3 = (const float*)d_in[8];
  float* out = (float*)d_out;

  const int N = in_sizes[0] / D;   // 50000
  const int E = in_sizes[1];       // 640000

  // Workspace layout (floats): [deg_out N][deg_in N][agg N*D][h N*D]
  float* ws = (float*)d_ws;
  float* inv_out = ws;
  float* inv_in = ws + N;
  float* agg = ws + 2 * (size_t)N;
  float* h = agg + (size_t)N * D;

  const int ZB = 2048;  // grid-stride zero blocks

  // Degrees + inv-sqrt normalization factors.
  gcn_zero_kernel<<<ZB, 256, 0, stream>>>(inv_out, 2 * (long long)N);
  gcn_degree_kernel<<<(E + 255) / 256, 256, 0, stream>>>(src, dst, inv_out, inv_in, E);
  gcn_invsqrt_kernel<<<(2 * N + 255) / 256, 256, 0, stream>>>(inv_out, 2 * N);

  const int spmm_blocks = (E * 32 + 255) / 256;
  const int gemm_blocks = N / 16;  // 50000 is a multiple of 16

  // Layer 1: feat -> h
  gcn_zero_kernel<<<ZB, 256, 0, stream>>>(agg, (long long)N * D);
  gcn_spmm_kernel<<<spmm_blocks, 256, 0, stream>>>(feat, src, dst, inv_out, agg, E);
  gcn_gemm_kernel<<<gemm_blocks, 256, 0, stream>>>(agg, inv_in, W1, b1, h, 1);

  // Layer 2: h -> h (spmm consumes h fully before gemm overwrites it)
  gcn_zero_kernel<<<ZB, 256, 0, stream>>>(agg, (long long)N * D);
  gcn_spmm_kernel<<<spmm_blocks, 256, 0, stream>>>(h, src, dst, inv_out, agg, E);
  gcn_gemm_kernel<<<gemm_blocks, 256, 0, stream>>>(agg, inv_in, W2, b2, h, 1);

  // Layer 3: h -> out (no relu)
  gcn_zero_kernel<<<ZB, 256, 0, stream>>>(agg, (long long)N * D);
  gcn_spmm_kernel<<<spmm_blocks, 256, 0, stream>>>(h, src, dst, inv_out, agg, E);
  gcn_gemm_kernel<<<gemm_blocks, 256, 0, stream>>>(agg, inv_in, W3, b3, out, 0);
}